// QuestAttention_3066606649969
// MI455X (gfx1250) — compile-verified
//
#include <hip/hip_runtime.h>
#include <math.h>

// ---- problem constants (match reference) ----
#define HID   4096
#define NH    32
#define NKV   32
#define DHEAD 128
#define KVLEN 32768
#define PAGE  16
#define NPAGE 2048          // KVLEN / PAGE
#define TOPK  256
#define SCALE 0.08838834764831843f   // 1/sqrt(128)

#define ISPLIT 8            // split-K for the GEMVs
#define NSPLIT 16           // flash-decoding splits per head (16 pages per block)
#define QKV_COLS (3*HID)

typedef float v2f __attribute__((ext_vector_type(2)));
typedef float v8f __attribute__((ext_vector_type(8)));

// ------------------------------------------------------------------
// 1) QKV projection, split-K GEMV: x(1x4096) @ [Wq|Wk|Wv]
//    grid (48, ISPLIT) x 256.  Coalesced over output column j,
//    x[i] is wave-uniform -> scalar loads.
// ------------------------------------------------------------------
__global__ void k_qkv_partial(const float* __restrict__ x,
                              const float* __restrict__ Wq,
                              const float* __restrict__ Wk,
                              const float* __restrict__ Wv,
                              float* __restrict__ part) {
  const int j = blockIdx.x * blockDim.x + threadIdx.x;   // 0..12287
  const int split = blockIdx.y;
  const float* W = (j < HID) ? Wq : ((j < 2*HID) ? Wk : Wv);
  const int col = j & (HID - 1);
  const int i0 = split * (HID / ISPLIT);
  float acc = 0.f;
#pragma unroll 4
  for (int i = 0; i < HID / ISPLIT; ++i)
    acc = fmaf(x[i0 + i], W[(size_t)(i0 + i) * HID + col], acc);
  part[(size_t)split * QKV_COLS + j] = acc;
}

__global__ void k_qkv_reduce(const float* __restrict__ part,
                             float* __restrict__ qkv) {
  const int j = blockIdx.x * blockDim.x + threadIdx.x;
  float s = 0.f;
#pragma unroll
  for (int sp = 0; sp < ISPLIT; ++sp) s += part[(size_t)sp * QKV_COLS + j];
  qkv[j] = s;
}

// ------------------------------------------------------------------
// 2) RoPE at pos = 32768 on q and k_new. grid(64) x 128.
//    Double-precision trig for the large-angle argument reduction.
// ------------------------------------------------------------------
__global__ void k_rope(const float* __restrict__ qkv,
                       float* __restrict__ q_r,
                       float* __restrict__ k_r) {
  const int b = blockIdx.x;
  const int tensor = b >> 5;          // 0 = q, 1 = k_new
  const int h = b & 31;
  const int d = threadIdx.x;          // 0..127
  const float* src = qkv + tensor * HID + h * DHEAD;
  float* dst = (tensor ? k_r : q_r) + h * DHEAD;
  const int jj = d & 63;
  const double ang = 32768.0 * pow(10000.0, -(double)(2 * jj) / 128.0);
  const double c = cos(ang), s = sin(ang);
  const float x1 = src[d];
  const float rot = (d < 64) ? -src[d + 64] : src[d - 64];
  dst[d] = (float)((double)x1 * c + (double)rot * s);
}

// ------------------------------------------------------------------
// 3) Fused page min/max + Quest score.  One wave per (page, head):
//    65536 waves streaming the full 512 MB k_cache with float4 loads.
//    grid(8192) x 256 (8 waves / block).
// ------------------------------------------------------------------
__global__ void k_est(const float* __restrict__ kc,
                      const float* __restrict__ q_r,
                      float* __restrict__ est) {
  const int g = blockIdx.x * (blockDim.x >> 5) + (threadIdx.x >> 5);
  const int lane = threadIdx.x & 31;
  const int h = g & (NKV - 1);
  const int p = g >> 5;
  float4 mn, mx;
  {
    const float4 v = reinterpret_cast<const float4*>(
        kc + ((size_t)(p * PAGE) * NKV + h) * DHEAD)[lane];
    mn = v; mx = v;
  }
#pragma unroll
  for (int t = 1; t < PAGE; ++t) {
    const float4 v = reinterpret_cast<const float4*>(
        kc + ((size_t)(p * PAGE + t) * NKV + h) * DHEAD)[lane];
    mn.x = fminf(mn.x, v.x); mn.y = fminf(mn.y, v.y);
    mn.z = fminf(mn.z, v.z); mn.w = fminf(mn.w, v.w);
    mx.x = fmaxf(mx.x, v.x); mx.y = fmaxf(mx.y, v.y);
    mx.z = fmaxf(mx.z, v.z); mx.w = fmaxf(mx.w, v.w);
  }
  const float4 q4 = reinterpret_cast<const float4*>(q_r + h * DHEAD)[lane];
  float e = fmaxf(q4.x * mn.x, q4.x * mx.x)
          + fmaxf(q4.y * mn.y, q4.y * mx.y)
          + fmaxf(q4.z * mn.z, q4.z * mx.z)
          + fmaxf(q4.w * mn.w, q4.w * mx.w);
#pragma unroll
  for (int off = 16; off; off >>= 1) e += __shfl_xor(e, off, 32);
  if (lane == 0) est[(size_t)h * NPAGE + p] = e;
}

// ------------------------------------------------------------------
// 4) Top-256 of 2048 per head: in-LDS bitonic sort (16 KB), 1 block/head.
// ------------------------------------------------------------------
__global__ void k_topk(const float* __restrict__ est,
                       int* __restrict__ topidx) {
  __shared__ unsigned long long keys[NPAGE];
  const int h = blockIdx.x, tid = threadIdx.x;
  for (int i = tid; i < NPAGE; i += blockDim.x) {
    unsigned u = __float_as_uint(est[(size_t)h * NPAGE + i]);
    u = (u & 0x80000000u) ? ~u : (u | 0x80000000u);   // order-preserving map
    keys[i] = ((unsigned long long)u << 32) | (unsigned)i;
  }
  for (unsigned k = 2; k <= NPAGE; k <<= 1)
    for (unsigned j = k >> 1; j; j >>= 1) {
      __syncthreads();
      for (unsigned i = tid; i < NPAGE; i += blockDim.x) {
        const unsigned ixj = i ^ j;
        if (ixj > i) {
          const unsigned long long a = keys[i], b = keys[ixj];
          const bool desc = ((i & k) == 0);
          if (desc ? (a < b) : (a > b)) { keys[i] = b; keys[ixj] = a; }
        }
      }
    }
  __syncthreads();
  for (int i = tid; i < TOPK; i += blockDim.x)
    topidx[h * TOPK + i] = (int)(keys[i] & 0xFFFFFFFFu);
}

// ------------------------------------------------------------------
// 5) Sparse attention, flash-decoding partials.
//    grid (NH, NSPLIT) x 128 (4 waves).  Each wave: 4 pages.
//    Scores via V_WMMA_F32_16X16X4_F32 on a padded LDS K-tile
//    (stride 132 -> conflict-free 64-bank fragment reads).
// ------------------------------------------------------------------
__global__ void k_attn_partial(const float* __restrict__ kc,
                               const float* __restrict__ vc,
                               const float* __restrict__ q_r,
                               const int* __restrict__ topidx,
                               float* __restrict__ part) {
  __shared__ float Kt[4 * 16 * 132];     // per-wave K tiles (reused for combine)
  __shared__ float qsm[DHEAD];
  const int h = blockIdx.x;
  const int sp = blockIdx.y;
  const int tid = threadIdx.x;
  const int w = tid >> 5;
  const int lane = tid & 31;
  qsm[tid & (DHEAD - 1)] = q_r[h * DHEAD + (tid & (DHEAD - 1))];
  __syncthreads();

  float* myK = Kt + w * (16 * 132);
  float m = -INFINITY, l = 0.f;
  float4 acc = make_float4(0.f, 0.f, 0.f, 0.f);
  const int koff = (lane < 16) ? 0 : 2;  // A/B fragment K-half (ISA 7.12.2)
  const int row = lane & 15;             // token row this lane owns in A

  for (int i = 0; i < 4; ++i) {
    const int page = topidx[h * TOPK + sp * 16 + w * 4 + i];
    // coalesced K tile -> LDS (16 rows x 128 floats, stride 132)
    for (int r = 0; r < PAGE; ++r) {
      const float4 kk = reinterpret_cast<const float4*>(
          kc + ((size_t)(page * PAGE + r) * NKV + h) * DHEAD)[lane];
      float* dp = myK + r * 132 + lane * 4;
      dp[0] = kk.x; dp[1] = kk.y; dp[2] = kk.z; dp[3] = kk.w;
    }
    // 16 token scores = K_tile(16x128) . q via 32 accumulated WMMAs
    v8f cacc = {0.f, 0.f, 0.f, 0.f, 0.f, 0.f, 0.f, 0.f};
#pragma unroll
    for (int c = 0; c < DHEAD; c += 4) {
      const float* kr = myK + row * 132 + c + koff;
      v2f a; a.x = kr[0];            a.y = kr[1];              // A: 16x4 tokens
      v2f b; b.x = qsm[c + koff];    b.y = qsm[c + koff + 1];  // B: q chunk, all N cols
      cacc = __builtin_amdgcn_wmma_f32_16x16x4_f32(
          false, a, false, b, (short)0, cacc, false, false);
    }
    // D[m][n] is column-independent: scores 0..7 live in lanes 0-15,
    // 8..15 in lanes 16-31 -> broadcast to the whole wave.
    float s[16];
#pragma unroll
    for (int t = 0; t < 8; ++t) {
      s[t]     = __shfl(cacc[t], 0, 32)  * SCALE;
      s[t + 8] = __shfl(cacc[t], 16, 32) * SCALE;
    }
    float pm = s[0];
#pragma unroll
    for (int t = 1; t < 16; ++t) pm = fmaxf(pm, s[t]);
    const float nm = fmaxf(m, pm);
    const float corr = expf(m - nm);
    l *= corr; acc.x *= corr; acc.y *= corr; acc.z *= corr; acc.w *= corr;
#pragma unroll
    for (int t = 0; t < PAGE; ++t) {
      const float wg = expf(s[t] - nm);
      l += wg;
      const float4 vv = reinterpret_cast<const float4*>(
          vc + ((size_t)(page * PAGE + t) * NKV + h) * DHEAD)[lane];
      acc.x = fmaf(wg, vv.x, acc.x); acc.y = fmaf(wg, vv.y, acc.y);
      acc.z = fmaf(wg, vv.z, acc.z); acc.w = fmaf(wg, vv.w, acc.w);
    }
    m = nm;
  }

  // cross-wave combine (reuse Kt as [4][130] = {m, l, u[128]})
  __syncthreads();
  float* cb = Kt;
  if (lane == 0) { cb[w * 130 + 0] = m; cb[w * 130 + 1] = l; }
  {
    float* up = cb + w * 130 + 2 + lane * 4;
    up[0] = acc.x; up[1] = acc.y; up[2] = acc.z; up[3] = acc.w;
  }
  __syncthreads();
  float M = cb[0];
#pragma unroll
  for (int ww = 1; ww < 4; ++ww) M = fmaxf(M, cb[ww * 130]);
  float L = 0.f, u = 0.f;
#pragma unroll
  for (int ww = 0; ww < 4; ++ww) {
    const float sc = expf(cb[ww * 130] - M);
    L += sc * cb[ww * 130 + 1];
    u += sc * cb[ww * 130 + 2 + tid];
  }
  float* pp = part + (size_t)(h * NSPLIT + sp) * 130;
  if (tid == 0) { pp[0] = M; pp[1] = L; }
  pp[2 + tid] = u;
}

// ------------------------------------------------------------------
// 6) Merge splits + the freshly projected token.  grid(NH) x 128.
// ------------------------------------------------------------------
__global__ void k_attn_combine(const float* __restrict__ part,
                               const float* __restrict__ q_r,
                               const float* __restrict__ k_r,
                               const float* __restrict__ v_new,
                               float* __restrict__ o_vec) {
  __shared__ float red[DHEAD];
  const int h = blockIdx.x, d = threadIdx.x;
  red[d] = q_r[h * DHEAD + d] * k_r[h * DHEAD + d];
  __syncthreads();
  for (int off = 64; off; off >>= 1) {
    if (d < off) red[d] += red[d + off];
    __syncthreads();
  }
  const float s_new = red[0] * SCALE;
  float M = s_new;
  for (int sp = 0; sp < NSPLIT; ++sp)
    M = fmaxf(M, part[(size_t)(h * NSPLIT + sp) * 130]);
  const float wn = expf(s_new - M);
  float L = wn;
  float o = wn * v_new[h * DHEAD + d];
  for (int sp = 0; sp < NSPLIT; ++sp) {
    const float* pp = part + (size_t)(h * NSPLIT + sp) * 130;
    const float sc = expf(pp[0] - M);
    L += sc * pp[1];
    o += sc * pp[2 + d];
  }
  o_vec[h * DHEAD + d] = o / L;
}

// ------------------------------------------------------------------
// 7) Output projection o(1x4096) @ Wo, split-K.
// ------------------------------------------------------------------
__global__ void k_outproj_partial(const float* __restrict__ o_vec,
                                  const float* __restrict__ Wo,
                                  float* __restrict__ part) {
  const int j = blockIdx.x * blockDim.x + threadIdx.x;  // 0..4095
  const int split = blockIdx.y;
  const int i0 = split * (HID / ISPLIT);
  float acc = 0.f;
#pragma unroll 4
  for (int i = 0; i < HID / ISPLIT; ++i)
    acc = fmaf(o_vec[i0 + i], Wo[(size_t)(i0 + i) * HID + j], acc);
  part[(size_t)split * HID + j] = acc;
}

__global__ void k_outproj_reduce(const float* __restrict__ part,
                                 float* __restrict__ out) {
  const int j = blockIdx.x * blockDim.x + threadIdx.x;
  float s = 0.f;
#pragma unroll
  for (int sp = 0; sp < ISPLIT; ++sp) s += part[(size_t)sp * HID + j];
  out[j] = s;
}

// ------------------------------------------------------------------
extern "C" void kernel_launch(void* const* d_in, const int* in_sizes, int n_in,
                              void* d_out, int out_size, void* d_ws, size_t ws_size,
                              hipStream_t stream) {
  const float* x  = (const float*)d_in[0];   // hidden_states (1,1,4096)
  const float* kc = (const float*)d_in[1];   // k_cache (32768,32,128)
  const float* vc = (const float*)d_in[2];   // v_cache
  const float* Wq = (const float*)d_in[3];
  const float* Wk = (const float*)d_in[4];
  const float* Wv = (const float*)d_in[5];
  const float* Wo = (const float*)d_in[6];
  float* out = (float*)d_out;

  // workspace layout (floats)
  float* ws        = (float*)d_ws;
  float* qkv_part  = ws;                               // ISPLIT * 12288
  float* qkv       = qkv_part + (size_t)ISPLIT * QKV_COLS;   // 12288
  float* q_r       = qkv + QKV_COLS;                   // 4096
  float* k_r       = q_r + HID;                        // 4096
  float* est       = k_r + HID;                        // 32*2048
  int*   topidx    = (int*)(est + (size_t)NH * NPAGE); // 32*256 ints
  float* attn_part = (float*)(topidx + NH * TOPK);     // 32*16*130
  float* o_vec     = attn_part + (size_t)NH * NSPLIT * 130;  // 4096
  float* out_part  = o_vec + HID;                      // ISPLIT * 4096
  const float* v_new = qkv + 2 * HID;

  k_qkv_partial<<<dim3(QKV_COLS / 256, ISPLIT), 256, 0, stream>>>(x, Wq, Wk, Wv, qkv_part);
  k_qkv_reduce<<<QKV_COLS / 256, 256, 0, stream>>>(qkv_part, qkv);
  k_rope<<<64, 128, 0, stream>>>(qkv, q_r, k_r);
  k_est<<<(NPAGE * NKV) / 8, 256, 0, stream>>>(kc, q_r, est);
  k_topk<<<NH, 256, 0, stream>>>(est, topidx);
  k_attn_partial<<<dim3(NH, NSPLIT), 128, 0, stream>>>(kc, vc, q_r, topidx, attn_part);
  k_attn_combine<<<NH, 128, 0, stream>>>(attn_part, q_r, k_r, v_new, o_vec);
  k_outproj_partial<<<dim3(HID / 256, ISPLIT), 256, 0, stream>>>(o_vec, Wo, out_part);
  k_outproj_reduce<<<HID / 256, 256, 0, stream>>>(out_part, out);
}